// ProbabilisticAttention_21543555956800
// MI455X (gfx1250) — compile-verified
//
#include <hip/hip_runtime.h>
#include <cstddef>

#define Hh 2048
#define Cc 64
#define BG 16
#define NTILE 128   // Hh / 16
#define LQLD 17     // padded LDS row stride (floats)

typedef float v2f __attribute__((ext_vector_type(2)));
typedef float v8f __attribute__((ext_vector_type(8)));

// Fragment loader for the 16x4 f32 A layout (and symmetric 4x16 B layout):
// lane L<16 : vgpr0 = X[kc+0][base+L], vgpr1 = X[kc+1][base+L]
// lane L>=16: vgpr0 = X[kc+2][base+L-16], vgpr1 = X[kc+3][base+L-16]
static __device__ __forceinline__ v2f frag_g(const float* p, int ld, int base,
                                             int kc, int half, int m) {
  const float* a = p + (size_t)(kc + 2 * half) * ld + base + m;
  v2f r;
  r.x = a[0];
  r.y = a[ld];
  return r;
}

static __device__ __forceinline__ v8f wmma4(v2f a, v2f b, v8f c) {
  return __builtin_amdgcn_wmma_f32_16x16x4_f32(false, a, false, b, (short)0, c,
                                               false, false);
}

// ---------------------------------------------------------------------------
// K1: bias[bg,h] = 0.5 * beta * sum_c mu[bg,c,h]^2
// ---------------------------------------------------------------------------
__global__ void k_bias(const float* __restrict__ mu,
                       const float* __restrict__ beta,
                       float* __restrict__ bias) {
  const int idx = blockIdx.x * 256 + threadIdx.x;  // 0 .. BG*Hh-1
  const int bg = idx >> 11;                        // Hh == 2048
  const int h = idx & (Hh - 1);
  const float* p = mu + (size_t)bg * Cc * Hh + h;
  float s = 0.0f;
#pragma unroll 4
  for (int c = 0; c < Cc; ++c) {
    const float x = p[(size_t)c * Hh];
    s = fmaf(x, x, s);
  }
  bias[idx] = 0.5f * beta[0] * s;
}

// ---------------------------------------------------------------------------
// K2: per (bg, i-tile) wave: streaming softmax stats over j
//     logits[i,j] = a*(q^T zeta)[i,j] + bias[j];  m = rowmax, s = sum exp
// ---------------------------------------------------------------------------
__global__ __launch_bounds__(32) void k_stats(
    const float* __restrict__ q, const float* __restrict__ zeta,
    const float* __restrict__ bias, const float* __restrict__ alpha,
    float* __restrict__ mrow, float* __restrict__ srow) {
  const int wid = blockIdx.x;
  const int bg = wid >> 7;
  const int i0 = (wid & 127) << 4;
  const float* qbg = q + (size_t)bg * Cc * Hh;
  const float* zbg = zeta + (size_t)bg * Cc * Hh;
  const float* bbg = bias + (size_t)bg * Hh;
  const float a = alpha[0];
  const int lane = threadIdx.x & 31;
  const int half = lane >> 4;
  const int n = lane & 15;

  // A fragments of q^T (M=i, K=c), persistent across the j loop
  v2f afr[16];
#pragma unroll
  for (int kb = 0; kb < 16; ++kb) afr[kb] = frag_g(qbg, Hh, i0, kb * 4, half, n);

  float rmax[8], rsum[8];
#pragma unroll
  for (int v = 0; v < 8; ++v) {
    rmax[v] = -3.0e38f;
    rsum[v] = 0.0f;
  }

  for (int jt = 0; jt < NTILE; ++jt) {
    const int j0 = jt << 4;
    v8f d = {0.f, 0.f, 0.f, 0.f, 0.f, 0.f, 0.f, 0.f};
#pragma unroll
    for (int kb = 0; kb < 16; ++kb) {
      v2f bfr = frag_g(zbg, Hh, j0, kb * 4, half, n);  // B: zeta (K=c, N=j)
      d = wmma4(afr[kb], bfr, d);
    }
    const float bn = bbg[j0 + n];
#pragma unroll
    for (int v = 0; v < 8; ++v) {
      float t = fmaf(a, d[v], bn);
      // reduce over the 16 columns held within this half-wave (row fixed)
      float tm = t;
      tm = fmaxf(tm, __shfl_xor(tm, 1, 32));
      tm = fmaxf(tm, __shfl_xor(tm, 2, 32));
      tm = fmaxf(tm, __shfl_xor(tm, 4, 32));
      tm = fmaxf(tm, __shfl_xor(tm, 8, 32));
      const float nm = fmaxf(rmax[v], tm);
      float es = __expf(t - nm);
      es += __shfl_xor(es, 1, 32);
      es += __shfl_xor(es, 2, 32);
      es += __shfl_xor(es, 4, 32);
      es += __shfl_xor(es, 8, 32);
      rsum[v] = rsum[v] * __expf(rmax[v] - nm) + es;
      rmax[v] = nm;
    }
  }
  if (n == 0) {
#pragma unroll
    for (int v = 0; v < 8; ++v) {
      const int row = i0 + v + 8 * half;
      mrow[(size_t)bg * Hh + row] = rmax[v];
      srow[(size_t)bg * Hh + row] = rsum[v];
    }
  }
}

// ---------------------------------------------------------------------------
// K3: per (bg, j-tile) wave: zeta_new[c,j] = (sum_i w[i,j] q[c,i]) / sum_i w
//     w recomputed from (zeta_old, m, s)
// ---------------------------------------------------------------------------
__global__ __launch_bounds__(32) void k_zeta(
    const float* __restrict__ q, const float* __restrict__ zold,
    const float* __restrict__ bias, const float* __restrict__ alpha,
    const float* __restrict__ mrow, const float* __restrict__ srow,
    float* __restrict__ znew) {
  __shared__ float lq[Cc * LQLD];
  __shared__ float lw[16 * 16];
  const int wid = blockIdx.x;
  const int bg = wid >> 7;
  const int j0 = (wid & 127) << 4;
  const float* qbg = q + (size_t)bg * Cc * Hh;
  const float* zbg = zold + (size_t)bg * Cc * Hh;
  const float* bbg = bias + (size_t)bg * Hh;
  const float* mbg = mrow + (size_t)bg * Hh;
  const float* sbg = srow + (size_t)bg * Hh;
  const float a = alpha[0];
  const int lane = threadIdx.x & 31;
  const int half = lane >> 4;
  const int n = lane & 15;

  // B fragments of zeta (K=c, N=j) for the fixed j-tile, persistent
  v2f bz[16];
#pragma unroll
  for (int kb = 0; kb < 16; ++kb) bz[kb] = frag_g(zbg, Hh, j0, kb * 4, half, n);
  const float bn = bbg[j0 + n];

  const v8f vz = {0.f, 0.f, 0.f, 0.f, 0.f, 0.f, 0.f, 0.f};
  v8f acc[4];
#pragma unroll
  for (int t = 0; t < 4; ++t) acc[t] = vz;
  float csum = 0.0f;

  for (int it = 0; it < NTILE; ++it) {
    const int i0 = it << 4;
    // stage q[c, i0:i0+16] tile into LDS (coalesced)
    for (int idx = lane; idx < Cc * 16; idx += 32) {
      const int c = idx >> 4, ii = idx & 15;
      lq[c * LQLD + ii] = qbg[(size_t)c * Hh + i0 + ii];
    }
    __syncthreads();
    // GEMM1: qz[i,j] = sum_c q[c,i] zeta[c,j]
    v8f d = vz;
#pragma unroll
    for (int kb = 0; kb < 16; ++kb) {
      const int kc = kb * 4;
      v2f afr;  // A: q^T (M=i, K=c) read from LDS
      afr.x = lq[(kc + 2 * half) * LQLD + n];
      afr.y = lq[(kc + 2 * half + 1) * LQLD + n];
      d = wmma4(afr, bz[kb], d);
    }
    // w tile -> LDS, plus column-sum accumulation
#pragma unroll
    for (int v = 0; v < 8; ++v) {
      const int row = i0 + v + 8 * half;
      const float w =
          __expf(fmaf(a, d[v], bn) - mbg[row]) / (sbg[row] + 1e-20f);
      lw[(v + 8 * half) * 16 + n] = w;
      csum += w;
    }
    __syncthreads();
    // GEMM2: acc[c,j] += q[c,i] * w[i,j]  (K = i, 4 steps of 4)
#pragma unroll
    for (int t = 0; t < 4; ++t) {
#pragma unroll
      for (int kb = 0; kb < 4; ++kb) {
        const int kc = kb * 4;
        v2f a2;  // A: q (M=c, K=i)
        a2.x = lq[(t * 16 + n) * LQLD + kc + 2 * half];
        a2.y = lq[(t * 16 + n) * LQLD + kc + 2 * half + 1];
        v2f b2;  // B: w (K=i, N=j)
        b2.x = lw[(kc + 2 * half) * 16 + n];
        b2.y = lw[(kc + 2 * half + 1) * 16 + n];
        acc[t] = wmma4(a2, b2, acc[t]);
      }
    }
    __syncthreads();
  }
  csum += __shfl_xor(csum, 16, 32);  // combine the two half-wave row groups
  const float inv = 1.0f / csum;
  float* zo = znew + (size_t)bg * Cc * Hh;
#pragma unroll
  for (int t = 0; t < 4; ++t) {
#pragma unroll
    for (int v = 0; v < 8; ++v) {
      const int c = t * 16 + v + 8 * half;
      zo[(size_t)c * Hh + j0 + n] = acc[t][v] * inv;
    }
  }
}

// ---------------------------------------------------------------------------
// K4: per (bg, i-tile) wave: out[c,i] = sum_j w[i,j] mu[c,j]
//     logits computed TRANSPOSED (M=j, N=i) so w^T lands in B orientation
// ---------------------------------------------------------------------------
__global__ __launch_bounds__(32) void k_value(
    const float* __restrict__ q, const float* __restrict__ zeta,
    const float* __restrict__ mu, const float* __restrict__ bias,
    const float* __restrict__ alpha, const float* __restrict__ mrow,
    const float* __restrict__ srow, float* __restrict__ out) {
  __shared__ float lmu[Cc * LQLD];
  __shared__ float lw[16 * 16];
  const int wid = blockIdx.x;
  const int bg = wid >> 7;
  const int i0 = (wid & 127) << 4;
  const float* qbg = q + (size_t)bg * Cc * Hh;
  const float* zbg = zeta + (size_t)bg * Cc * Hh;
  const float* mubg = mu + (size_t)bg * Cc * Hh;
  const float* bbg = bias + (size_t)bg * Hh;
  const float a = alpha[0];
  const int lane = threadIdx.x & 31;
  const int half = lane >> 4;
  const int n = lane & 15;

  // B fragments of q (K=c, N=i) for the fixed i-tile, persistent
  v2f bq[16];
#pragma unroll
  for (int kb = 0; kb < 16; ++kb) bq[kb] = frag_g(qbg, Hh, i0, kb * 4, half, n);
  const float mr = mrow[(size_t)bg * Hh + i0 + n];
  const float sr = srow[(size_t)bg * Hh + i0 + n] + 1e-20f;

  const v8f vz = {0.f, 0.f, 0.f, 0.f, 0.f, 0.f, 0.f, 0.f};
  v8f acc[4];
#pragma unroll
  for (int t = 0; t < 4; ++t) acc[t] = vz;

  for (int jt = 0; jt < NTILE; ++jt) {
    const int j0 = jt << 4;
    // stage mu[c, j0:j0+16] tile into LDS
    for (int idx = lane; idx < Cc * 16; idx += 32) {
      const int c = idx >> 4, jj = idx & 15;
      lmu[c * LQLD + jj] = mubg[(size_t)c * Hh + j0 + jj];
    }
    __syncthreads();
    // transposed logits: tz[j,i] = sum_c zeta[c,j] q[c,i]
    v8f d = vz;
#pragma unroll
    for (int kb = 0; kb < 16; ++kb) {
      v2f az = frag_g(zbg, Hh, j0, kb * 4, half, n);  // A: zeta^T (M=j, K=c)
      d = wmma4(az, bq[kb], d);
    }
    // w^T[j,i] -> LDS ; bias by row j, softmax stats by column i
#pragma unroll
    for (int v = 0; v < 8; ++v) {
      const float bj = bbg[j0 + v + 8 * half];
      lw[(v + 8 * half) * 16 + n] = __expf(fmaf(a, d[v], bj) - mr) / sr;
    }
    __syncthreads();
    // GEMM2: acc[c,i] += mu[c,j] * w^T[j,i]  (K = j, 4 steps of 4)
#pragma unroll
    for (int t = 0; t < 4; ++t) {
#pragma unroll
      for (int kb = 0; kb < 4; ++kb) {
        const int kc = kb * 4;
        v2f a2;  // A: mu (M=c, K=j)
        a2.x = lmu[(t * 16 + n) * LQLD + kc + 2 * half];
        a2.y = lmu[(t * 16 + n) * LQLD + kc + 2 * half + 1];
        v2f b2;  // B: w^T (K=j, N=i)
        b2.x = lw[(kc + 2 * half) * 16 + n];
        b2.y = lw[(kc + 2 * half + 1) * 16 + n];
        acc[t] = wmma4(a2, b2, acc[t]);
      }
    }
    __syncthreads();
  }
  float* obg = out + (size_t)bg * Cc * Hh;
#pragma unroll
  for (int t = 0; t < 4; ++t) {
#pragma unroll
    for (int v = 0; v < 8; ++v) {
      const int c = t * 16 + v + 8 * half;
      obg[(size_t)c * Hh + i0 + n] = acc[t][v];
    }
  }
}

// ---------------------------------------------------------------------------
extern "C" void kernel_launch(void* const* d_in, const int* in_sizes, int n_in,
                              void* d_out, int out_size, void* d_ws,
                              size_t ws_size, hipStream_t stream) {
  const float* q = (const float*)d_in[0];
  const float* zeta = (const float*)d_in[1];
  const float* alpha = (const float*)d_in[2];
  const float* mu = (const float*)d_in[3];
  const float* beta = (const float*)d_in[4];
  float* out = (float*)d_out;

  float* ws = (float*)d_ws;
  float* bias = ws;                          // BG*Hh floats
  float* mrow = ws + (size_t)BG * Hh;        // BG*Hh
  float* srow = ws + (size_t)2 * BG * Hh;    // BG*Hh
  float* za = ws + (size_t)3 * BG * Hh;      // BG*Cc*Hh
  float* zb = za + (size_t)BG * Cc * Hh;     // BG*Cc*Hh  (~17 MB total)

  const dim3 wgrid(BG * NTILE), wblk(32);

  k_bias<<<dim3((BG * Hh) / 256), dim3(256), 0, stream>>>(mu, beta, bias);
  // w0 = UW(zeta_in)
  k_stats<<<wgrid, wblk, 0, stream>>>(q, zeta, bias, alpha, mrow, srow);
  // iter 1: zeta_a = Z(w0);  w1 = UW(zeta_a)
  k_zeta<<<wgrid, wblk, 0, stream>>>(q, zeta, bias, alpha, mrow, srow, za);
  k_stats<<<wgrid, wblk, 0, stream>>>(q, za, bias, alpha, mrow, srow);
  // iter 2: zeta_b = Z(w1);  w2 = UW(zeta_b)
  k_zeta<<<wgrid, wblk, 0, stream>>>(q, za, bias, alpha, mrow, srow, zb);
  k_stats<<<wgrid, wblk, 0, stream>>>(q, zb, bias, alpha, mrow, srow);
  // out = V(w2)
  k_value<<<wgrid, wblk, 0, stream>>>(q, zb, mu, bias, alpha, mrow, srow, out);

  (void)in_sizes;
  (void)n_in;
  (void)out_size;
  (void)ws_size;
}